// Net_39230231281891
// MI455X (gfx1250) — compile-verified
//
#include <hip/hip_runtime.h>
#include <hip/hip_bf16.h>
#include <math.h>

// ---------------------------------------------------------------------------
// 2-layer GCN (PyG GCNConv x2 + log_softmax) for MI455X / gfx1250.
//  x:[N,512] f32, edge_index:[2,E] int32, W1:[512,16], b1:[16], W2:[16,40],
//  b2:[40]  ->  out:[N,40] f32 log-softmax.
// GEMMs: V_WMMA_F32_16X16X4_F32 (full fp32 precision on the matrix pipe).
// Edge propagation: norm-weighted gather + float atomic scatter-add; hot
// buffers (xw 6.4MB, hw 16MB, accumulators) stay L2-resident (192MB L2).
// ---------------------------------------------------------------------------

typedef float v2f __attribute__((ext_vector_type(2)));
typedef float v8f __attribute__((ext_vector_type(8)));

static constexpr int F_IN_C = 512;
static constexpr int HID_C  = 16;
static constexpr int NCLS_C = 40;
static constexpr int GEMM_WAVES = 4;     // waves (tiles) per block

// ---- degree count (self-loop added later as +1) ---------------------------
__global__ __launch_bounds__(256)
void k_count_deg(const int* __restrict__ dst, unsigned* __restrict__ cnt, int e) {
  int t = blockIdx.x * blockDim.x + threadIdx.x;
  if (t < e) atomicAdd(&cnt[dst[t]], 1u);
}

__global__ __launch_bounds__(256)
void k_dinv(const unsigned* __restrict__ cnt, float* __restrict__ dinv, int n) {
  int t = blockIdx.x * blockDim.x + threadIdx.x;
  if (t < n) dinv[t] = rsqrtf((float)cnt[t] + 1.0f);   // deg >= 1 always
}

__global__ __launch_bounds__(256)
void k_norm(const int* __restrict__ src, const int* __restrict__ dst,
            const float* __restrict__ dinv, float* __restrict__ norm, int e) {
  int t = blockIdx.x * blockDim.x + threadIdx.x;
  if (t < e) norm[t] = dinv[src[t]] * dinv[dst[t]];
}

// ---- GEMM1: xw = x @ W1   [N,512]x[512,16], f32 WMMA 16x16x4 --------------
__global__ __launch_bounds__(GEMM_WAVES * 32)
void k_gemm1_wmma(const float* __restrict__ x, const float* __restrict__ W1,
                  float* __restrict__ xw, int n) {
  const int lane = threadIdx.x & 31;
  const int wave = threadIdx.x >> 5;
  const int m0   = (blockIdx.x * GEMM_WAVES + wave) * 16;
  if (m0 >= n) return;                     // whole-wave uniform exit
  const int lrow = lane & 15;
  const int kb   = (lane >> 4) * 2;        // 0 or 2 (K pair within 4-step)
  int mrow = m0 + lrow; if (mrow >= n) mrow = n - 1;   // clamp: EXEC stays full
  const float* __restrict__ xr = x + (long long)mrow * F_IN_C;

  v8f c = {0.f, 0.f, 0.f, 0.f, 0.f, 0.f, 0.f, 0.f};
  for (int k0 = 0; k0 < F_IN_C; k0 += 4) {
    v2f a, b;
    a.x = xr[k0 + kb];
    a.y = xr[k0 + kb + 1];
    b.x = W1[(k0 + kb)     * HID_C + lrow];
    b.y = W1[(k0 + kb + 1) * HID_C + lrow];
    c = __builtin_amdgcn_wmma_f32_16x16x4_f32(false, a, false, b,
                                              (short)0, c, false, false);
  }
  // C/D layout: VGPR r -> row (r + (lane>>4)*8), col lane&15
  const int rbase = m0 + ((lane >> 4) << 3);
  float* __restrict__ o = xw + (long long)rbase * HID_C + lrow;
  if (m0 + 16 <= n) {                      // full tile: unguarded stores
#pragma unroll
    for (int r = 0; r < 8; ++r) o[(long long)r * HID_C] = c[r];
  } else {                                 // partial tile (never hit for N=100000)
#pragma unroll
    for (int r = 0; r < 8; ++r)
      if (rbase + r < n) o[(long long)r * HID_C] = c[r];
  }
}

// ---- GEMM2: hw = h @ W2   [N,16]x[16,40], 3 column tiles ------------------
__global__ __launch_bounds__(GEMM_WAVES * 32)
void k_gemm2_wmma(const float* __restrict__ h, const float* __restrict__ W2,
                  float* __restrict__ hw, int n) {
  const int lane = threadIdx.x & 31;
  const int wave = threadIdx.x >> 5;
  const int m0   = (blockIdx.x * GEMM_WAVES + wave) * 16;
  if (m0 >= n) return;
  const int lrow = lane & 15;
  const int kb   = (lane >> 4) * 2;
  int mrow = m0 + lrow; if (mrow >= n) mrow = n - 1;
  const float* __restrict__ hr = h + (long long)mrow * HID_C;

  v8f c0 = {0.f,0.f,0.f,0.f,0.f,0.f,0.f,0.f};
  v8f c1 = c0, c2 = c0;
  const bool col2ok = lrow < (NCLS_C - 32);   // third tile: cols 32..39 valid
  const int  col2   = 32 + lrow;
#pragma unroll
  for (int k0 = 0; k0 < HID_C; k0 += 4) {
    v2f a, b0, b1, b2;
    a.x = hr[k0 + kb];
    a.y = hr[k0 + kb + 1];
    const float* w0 = W2 + (k0 + kb)     * NCLS_C;
    const float* w1 = W2 + (k0 + kb + 1) * NCLS_C;
    b0.x = w0[lrow];       b0.y = w1[lrow];
    b1.x = w0[16 + lrow];  b1.y = w1[16 + lrow];
    b2.x = col2ok ? w0[col2] : 0.f;
    b2.y = col2ok ? w1[col2] : 0.f;
    c0 = __builtin_amdgcn_wmma_f32_16x16x4_f32(false, a, false, b0, (short)0, c0, false, false);
    c1 = __builtin_amdgcn_wmma_f32_16x16x4_f32(false, a, false, b1, (short)0, c1, false, false);
    c2 = __builtin_amdgcn_wmma_f32_16x16x4_f32(false, a, false, b2, (short)0, c2, false, false);
  }
  const int rbase = m0 + ((lane >> 4) << 3);
  if (m0 + 16 <= n) {
#pragma unroll
    for (int r = 0; r < 8; ++r) {
      float* o = hw + (long long)(rbase + r) * NCLS_C;
      o[lrow]      = c0[r];
      o[16 + lrow] = c1[r];
      if (col2ok) o[col2] = c2[r];
    }
  } else {
#pragma unroll
    for (int r = 0; r < 8; ++r) {
      if (rbase + r < n) {
        float* o = hw + (long long)(rbase + r) * NCLS_C;
        o[lrow]      = c0[r];
        o[16 + lrow] = c1[r];
        if (col2ok) o[col2] = c2[r];
      }
    }
  }
}

// ---- edge scatter-add: agg[dst] += norm[e] * feat[src] --------------------
template <int FD>
__global__ __launch_bounds__(256)
void k_scatter(const int* __restrict__ src, const int* __restrict__ dst,
               const float* __restrict__ norm, const float* __restrict__ feat,
               float* __restrict__ agg, int e) {
  long long t = (long long)blockIdx.x * blockDim.x + threadIdx.x;
  if (t >= (long long)e * FD) return;
  int ei = (int)(t / FD);
  int f  = (int)(t - (long long)ei * FD);
  int s = src[ei], d = dst[ei];
  atomicAdd(&agg[(long long)d * FD + f], norm[ei] * feat[(long long)s * FD + f]);
}

// ---- layer1 finalize: h = relu(agg1 + dinv^2*xw + b1), in place -----------
__global__ __launch_bounds__(256)
void k_finalize1(float* __restrict__ agg1, const float* __restrict__ xw,
                 const float* __restrict__ dinv, const float* __restrict__ b1,
                 int n) {
  int t = blockIdx.x * blockDim.x + threadIdx.x;
  if (t >= n * HID_C) return;
  int i = t >> 4, f = t & 15;
  float di = dinv[i];
  float v = agg1[t] + di * di * xw[t] + b1[f];
  agg1[t] = fmaxf(v, 0.0f);
}

// ---- layer2 finalize + log_softmax, in place on d_out ---------------------
__global__ __launch_bounds__(256)
void k_logsoftmax(float* __restrict__ out, const float* __restrict__ hw,
                  const float* __restrict__ dinv, const float* __restrict__ b2,
                  int n) {
  int i = blockIdx.x * blockDim.x + threadIdx.x;
  if (i >= n) return;
  float di = dinv[i], sw = di * di;
  float v[NCLS_C];
  float mx = -INFINITY;
  long long base = (long long)i * NCLS_C;
#pragma unroll
  for (int c = 0; c < NCLS_C; ++c) {
    float t = out[base + c] + sw * hw[base + c] + b2[c];
    v[c] = t;
    mx = fmaxf(mx, t);
  }
  float s = 0.f;
#pragma unroll
  for (int c = 0; c < NCLS_C; ++c) s += expf(v[c] - mx);
  float lse = mx + logf(s);
#pragma unroll
  for (int c = 0; c < NCLS_C; ++c) out[base + c] = v[c] - lse;
}

// ---------------------------------------------------------------------------
extern "C" void kernel_launch(void* const* d_in, const int* in_sizes, int n_in,
                              void* d_out, int out_size, void* d_ws, size_t ws_size,
                              hipStream_t stream) {
  (void)n_in; (void)out_size; (void)ws_size;
  const float* x  = (const float*)d_in[0];
  const float* W1 = (const float*)d_in[1];
  const float* b1 = (const float*)d_in[2];
  const float* W2 = (const float*)d_in[3];
  const float* b2 = (const float*)d_in[4];
  const int*  eix = (const int*)d_in[5];      // int32 per harness bridging

  const int n = in_sizes[0] / F_IN_C;         // 100000
  const int e = in_sizes[5] / 2;              // 3200000
  const int* src = eix;
  const int* dst = eix + e;

  // workspace carve-up (~42 MB)
  char* ws = (char*)d_ws;
  size_t off = 0;
  auto take = [&](size_t bytes) {
    char* p = ws + off;
    off += (bytes + 255) & ~size_t(255);
    return p;
  };
  unsigned* cnt  = (unsigned*)take((size_t)n * 4);
  float*    dinv = (float*)   take((size_t)n * 4);
  float*    norm = (float*)   take((size_t)e * 4);
  float*    xw   = (float*)   take((size_t)n * HID_C * 4);
  float*    agg1 = (float*)   take((size_t)n * HID_C * 4);   // becomes h
  float*    hw   = (float*)   take((size_t)n * NCLS_C * 4);
  float*    out  = (float*)d_out;                             // layer-2 accumulator

  hipMemsetAsync(cnt,  0, (size_t)n * 4, stream);
  hipMemsetAsync(agg1, 0, (size_t)n * HID_C * 4, stream);
  hipMemsetAsync(out,  0, (size_t)n * NCLS_C * 4, stream);

  k_count_deg<<<(e + 255) / 256, 256, 0, stream>>>(dst, cnt, e);
  k_dinv<<<(n + 255) / 256, 256, 0, stream>>>(cnt, dinv, n);
  k_norm<<<(e + 255) / 256, 256, 0, stream>>>(src, dst, dinv, norm, e);

  const int tiles = (n + 15) / 16;
  const int gblocks = (tiles + GEMM_WAVES - 1) / GEMM_WAVES;
  k_gemm1_wmma<<<gblocks, GEMM_WAVES * 32, 0, stream>>>(x, W1, xw, n);

  {
    long long tot = (long long)e * HID_C;
    k_scatter<HID_C><<<(unsigned)((tot + 255) / 256), 256, 0, stream>>>(
        src, dst, norm, xw, agg1, e);
  }
  k_finalize1<<<(n * HID_C + 255) / 256, 256, 0, stream>>>(agg1, xw, dinv, b1, n);

  k_gemm2_wmma<<<gblocks, GEMM_WAVES * 32, 0, stream>>>(agg1, W2, hw, n);

  {
    long long tot = (long long)e * NCLS_C;
    k_scatter<NCLS_C><<<(unsigned)((tot + 255) / 256), 256, 0, stream>>>(
        src, dst, norm, hw, out, e);
  }
  k_logsoftmax<<<(n + 255) / 256, 256, 0, stream>>>(out, hw, dinv, b2, n);
}